// GraphConvolution_31061203485065
// MI455X (gfx1250) — compile-verified
//
#include <hip/hip_runtime.h>
#include <math.h>

typedef __attribute__((ext_vector_type(2))) float v2f;
typedef __attribute__((ext_vector_type(4))) float v4f;
typedef __attribute__((ext_vector_type(8))) float v8f;

#define D 128
#define LN_EPS 1e-5f

// ---------------------------------------------------------------------------
// Kernel 1: h = features @ weight   (fp32 WMMA 16x16x4)
// One block = 256 threads = 8 waves. Each wave computes a 16-row x 128-col
// strip of H. Weight (128x128 f32 = 64KB) is staged in LDS once per block.
// ---------------------------------------------------------------------------
__global__ __launch_bounds__(256) void gcn_gemm_wmma(
    const float* __restrict__ A,   // [N,128] features
    const float* __restrict__ W,   // [128,128] weight
    float* __restrict__ H,         // [N,128] out
    int N) {
  __shared__ __align__(16) float lw[D * D];  // 64 KB (WGP has 320 KB)

  const int tid = threadIdx.x;
  // Cooperative load of the full weight matrix into LDS (float4 strides).
  for (int i = tid * 4; i < D * D; i += 256 * 4) {
    *(v4f*)(&lw[i]) = *(const v4f*)(&W[i]);
  }
  __syncthreads();

  const int wave = tid >> 5;
  const int lane = tid & 31;
  const int laneM = lane & 15;     // A row / B,C column within 16-wide tile
  const int laneH = lane >> 4;     // half selector: K-pairs {0,1} vs {2,3}

  const int rowTile = blockIdx.x * 128 + wave * 16;

  // Clamp A row for tail blocks: WMMA requires EXEC all-ones, so we load a
  // valid (duplicated) row and simply mask the stores below.
  int arowIdx = rowTile + laneM;
  if (arowIdx >= N) arowIdx = N - 1;
  const float* arow = A + (size_t)arowIdx * D;

  v8f acc[8];
#pragma unroll
  for (int nt = 0; nt < 8; ++nt) acc[nt] = (v8f)(0.0f);

  // K loop: 128 / 4 = 32 steps; 8 WMMAs per step (one per 16-col tile).
  for (int kk = 0; kk < D; kk += 4) {
    // A fragment (16x4 f32): lane holds K = kk + laneH*2 + {0,1} of row laneM
    v2f a = *(const v2f*)(arow + kk + laneH * 2);
    const int kb = kk + laneH * 2;
#pragma unroll
    for (int nt = 0; nt < 8; ++nt) {
      // B fragment (4x16 f32): VGPR0 = rows {kb}, VGPR1 = rows {kb+1},
      // N = nt*16 + laneM (mirrors the A half-split layout).
      v2f b;
      b.x = lw[kb * D + nt * 16 + laneM];
      b.y = lw[(kb + 1) * D + nt * 16 + laneM];
      acc[nt] = __builtin_amdgcn_wmma_f32_16x16x4_f32(
          /*neg_a=*/false, a, /*neg_b=*/false, b,
          /*c_mod=*/(short)0, acc[nt], /*reuse_a=*/false, /*reuse_b=*/false);
    }
  }

  // C/D layout: VGPR j -> M = j + laneH*8, N = nt*16 + laneM.
#pragma unroll
  for (int nt = 0; nt < 8; ++nt) {
#pragma unroll
    for (int j = 0; j < 8; ++j) {
      int m = rowTile + j + laneH * 8;
      if (m < N) H[(size_t)m * D + nt * 16 + laneM] = acc[nt][j];
    }
  }
}

// ---------------------------------------------------------------------------
// Kernel 2: zero the aggregation buffer (float4 grid-stride).
// ---------------------------------------------------------------------------
__global__ void gcn_zero_f4(float* __restrict__ p, long long n4) {
  long long i = (long long)blockIdx.x * blockDim.x + threadIdx.x;
  const long long stride = (long long)gridDim.x * blockDim.x;
  v4f z = (v4f)(0.0f);
  for (; i < n4; i += stride) ((v4f*)p)[i] = z;
}

// ---------------------------------------------------------------------------
// Kernel 3: SpMM scatter. One wave per edge; each lane owns 4 features.
// agg[row] += val * h[col]. h and agg are L2-resident (51 MB each, 192 MB L2).
// ---------------------------------------------------------------------------
__global__ __launch_bounds__(256) void gcn_spmm_scatter(
    const long long* __restrict__ idx,   // [2,E] int64 (row ; col)
    const float* __restrict__ vals,      // [E]
    const float* __restrict__ H,         // [N,128]
    float* __restrict__ agg,             // [N,128]
    int E) {
  const int lane = threadIdx.x & 31;
  int e = blockIdx.x * (blockDim.x >> 5) + (threadIdx.x >> 5);
  const int stride = gridDim.x * (blockDim.x >> 5);
  for (; e < E; e += stride) {
    const int r = (int)idx[e];
    const int c = (int)idx[(size_t)E + e];
    const float v = vals[e];
    v4f hv = *(const v4f*)(H + (size_t)c * D + lane * 4);
    float* dst = agg + (size_t)r * D + lane * 4;
    atomicAdd(dst + 0, v * hv.x);
    atomicAdd(dst + 1, v * hv.y);
    atomicAdd(dst + 2, v * hv.z);
    atomicAdd(dst + 3, v * hv.w);
  }
}

// ---------------------------------------------------------------------------
// Kernel 4: bias + ELU + LayerNorm. One wave32 per row, lane holds 4 feats.
// ---------------------------------------------------------------------------
__global__ __launch_bounds__(256) void gcn_epilogue(
    const float* __restrict__ agg, const float* __restrict__ bias,
    const float* __restrict__ gamma, const float* __restrict__ beta,
    float* __restrict__ out, int N) {
  const int lane = threadIdx.x & 31;
  int r = blockIdx.x * (blockDim.x >> 5) + (threadIdx.x >> 5);
  const int stride = gridDim.x * (blockDim.x >> 5);

  const v4f bv = *(const v4f*)(bias + lane * 4);
  const v4f gv = *(const v4f*)(gamma + lane * 4);
  const v4f bev = *(const v4f*)(beta + lane * 4);

  for (; r < N; r += stride) {
    v4f x = *(const v4f*)(agg + (size_t)r * D + lane * 4);
    float x0 = x.x + bv.x, x1 = x.y + bv.y, x2 = x.z + bv.z, x3 = x.w + bv.w;
    // ELU (alpha=1): x>0 ? x : expm1(x)
    x0 = x0 > 0.0f ? x0 : expm1f(x0);
    x1 = x1 > 0.0f ? x1 : expm1f(x1);
    x2 = x2 > 0.0f ? x2 : expm1f(x2);
    x3 = x3 > 0.0f ? x3 : expm1f(x3);

    float s = x0 + x1 + x2 + x3;
#pragma unroll
    for (int off = 16; off > 0; off >>= 1) s += __shfl_xor(s, off, 32);
    const float mu = s * (1.0f / 128.0f);

    const float d0 = x0 - mu, d1 = x1 - mu, d2 = x2 - mu, d3 = x3 - mu;
    float vs = d0 * d0 + d1 * d1 + d2 * d2 + d3 * d3;
#pragma unroll
    for (int off = 16; off > 0; off >>= 1) vs += __shfl_xor(vs, off, 32);
    const float rstd = rsqrtf(vs * (1.0f / 128.0f) + LN_EPS);

    v4f o;
    o.x = d0 * rstd * gv.x + bev.x;
    o.y = d1 * rstd * gv.y + bev.y;
    o.z = d2 * rstd * gv.z + bev.z;
    o.w = d3 * rstd * gv.w + bev.w;
    *(v4f*)(out + (size_t)r * D + lane * 4) = o;
  }
}

// ---------------------------------------------------------------------------
extern "C" void kernel_launch(void* const* d_in, const int* in_sizes, int n_in,
                              void* d_out, int out_size, void* d_ws, size_t ws_size,
                              hipStream_t stream) {
  const float* features       = (const float*)d_in[0];
  const long long* adj_indices = (const long long*)d_in[1];
  const float* adj_values     = (const float*)d_in[2];
  const float* weight         = (const float*)d_in[3];
  const float* bias           = (const float*)d_in[4];
  const float* ln_gamma       = (const float*)d_in[5];
  const float* ln_beta        = (const float*)d_in[6];

  const int N = in_sizes[0] / D;   // 100000
  const int E = in_sizes[2];       // 1600000

  float* H   = (float*)d_ws;                   // [N,128]  51.2 MB
  float* agg = H + (size_t)N * D;              // [N,128]  51.2 MB

  // 1) GEMM: 128 rows per block, 8 waves x 16-row strips.
  const int gemmBlocks = (N + 127) / 128;
  hipLaunchKernelGGL(gcn_gemm_wmma, dim3(gemmBlocks), dim3(256), 0, stream,
                     features, weight, H, N);

  // 2) Zero agg.
  const long long n4 = (long long)N * (D / 4);
  int zblocks = (int)((n4 + 255) / 256);
  if (zblocks > 8192) zblocks = 8192;
  hipLaunchKernelGGL(gcn_zero_f4, dim3(zblocks), dim3(256), 0, stream,
                     agg, n4);

  // 3) Edge scatter: one wave per edge (8 waves/block).
  const int sblocks = (E + 7) / 8;
  hipLaunchKernelGGL(gcn_spmm_scatter, dim3(sblocks), dim3(256), 0, stream,
                     adj_indices, adj_values, H, agg, E);

  // 4) Epilogue: one wave per row.
  const int eblocks = (N + 7) / 8;
  hipLaunchKernelGGL(gcn_epilogue, dim3(eblocks), dim3(256), 0, stream,
                     agg, bias, ln_gamma, ln_beta, (float*)d_out, N);
}